// HealpixConv_11295763988666
// MI455X (gfx1250) — compile-verified
//
#include <hip/hip_runtime.h>

// HEALPix graph conv: y[b,n,o] = sum_{k,c} w[o,k,c] * x_pad[b, neigh[n,k], c] + bias[o]
// GEMM view per 16-pixel tile: [16 x 144] * [144 x 16] via V_WMMA_F32_16X16X4_F32 (36 K-steps).

typedef __attribute__((ext_vector_type(2))) float v2f;
typedef __attribute__((ext_vector_type(8))) float v8f;

#define HC_NPIX 786432
#define HC_K    9
#define HC_CIN  16
#define HC_COUT 16
#define HC_WSZ  (HC_COUT * HC_K * HC_CIN)   // 2304 floats = 9216 B
#define HC_WAVES 8
#define HC_BLOCK (HC_WAVES * 32)

__global__ __launch_bounds__(HC_BLOCK) void healpix_conv_wmma_f32(
    const float* __restrict__ x,      // (2, NPIX, 16)
    const int*   __restrict__ neigh,  // (NPIX, 9), values in [0, NPIX] (NPIX == zero pad)
    const float* __restrict__ w,      // (16, 9, 16) row-major
    const float* __restrict__ bias,   // (16)
    float*       __restrict__ out)    // (2, NPIX, 16)
{
  __shared__ float wl[HC_WSZ];
  for (int i = threadIdx.x; i < HC_WSZ; i += HC_BLOCK) wl[i] = w[i];
  __syncthreads();

  const int lane = threadIdx.x & 31;
  const int wv   = threadIdx.x >> 5;
  const int m    = lane & 15;   // A: row M / B: col N / D: col N
  const int half = lane >> 4;   // selects K=2,3 (A/B) and M+=8 (D)

  const int tile = blockIdx.x * HC_WAVES + wv;
  const int p0   = tile << 4;
  if (p0 >= HC_NPIX) return;    // wave-uniform

  const int   pix = p0 + m;
  const float bo  = bias[m];

  const float* __restrict__ x0 = x;
  const float* __restrict__ x1 = x + (size_t)HC_NPIX * HC_CIN;  // batch 1

  v8f acc0 = {};
  v8f acc1 = {};

  #pragma unroll
  for (int k = 0; k < HC_K; ++k) {
    // neighbour index: streaming read-once -> non-temporal (keep L2 for x)
    const int idx = __builtin_nontemporal_load(&neigh[pix * HC_K + k]);
    const unsigned uidx  = (unsigned)idx;
    const bool     valid = uidx < (unsigned)HC_NPIX;   // idx==NPIX -> zero pad row
    const size_t   roff  = (size_t)(valid ? uidx : 0u) * HC_CIN;
    const float* r0 = x0 + roff;
    const float* r1 = x1 + roff;
    const v2f zz = {0.f, 0.f};

    #pragma unroll
    for (int g = 0; g < 4; ++g) {
      const int c = g * 4 + 2 * half;           // K-element pair this lane feeds
      v2f a0 = *(const v2f*)(r0 + c);
      v2f a1 = *(const v2f*)(r1 + c);
      a0 = valid ? a0 : zz;                     // branchless: EXEC stays all-ones
      a1 = valid ? a1 : zz;
      // B[kk][n] = w[n][k][c0+kk]; lane n=m supplies kk = 2*half, 2*half+1
      const v2f bv = *(const v2f*)(&wl[m * (HC_K * HC_CIN) + k * HC_CIN + c]);

      acc0 = __builtin_amdgcn_wmma_f32_16x16x4_f32(
                 false, a0, false, bv, (short)0, acc0, false, false);
      acc1 = __builtin_amdgcn_wmma_f32_16x16x4_f32(
                 false, a1, false, bv, (short)0, acc1, false, false);
    }
  }

  // D layout: VGPR v holds (row = v + 8*half, col = m). Output written once -> NT stores.
  float* o0 = out + (size_t)p0 * HC_COUT;
  float* o1 = out + (size_t)HC_NPIX * HC_COUT + (size_t)p0 * HC_COUT;
  #pragma unroll
  for (int v = 0; v < 8; ++v) {
    const int row = v + 8 * half;
    __builtin_nontemporal_store(acc0[v] + bo, &o0[row * HC_COUT + m]);
    __builtin_nontemporal_store(acc1[v] + bo, &o1[row * HC_COUT + m]);
  }
}

extern "C" void kernel_launch(void* const* d_in, const int* in_sizes, int n_in,
                              void* d_out, int out_size, void* d_ws, size_t ws_size,
                              hipStream_t stream) {
  (void)in_sizes; (void)n_in; (void)out_size; (void)d_ws; (void)ws_size;
  const float* x     = (const float*)d_in[0];
  const int*   neigh = (const int*)d_in[1];
  const float* w     = (const float*)d_in[2];
  const float* b     = (const float*)d_in[3];
  float*       out   = (float*)d_out;

  const int tiles  = HC_NPIX / 16;          // 49152
  const int blocks = tiles / HC_WAVES;      // 6144 blocks of 8 waves
  healpix_conv_wmma_f32<<<blocks, HC_BLOCK, 0, stream>>>(x, neigh, w, b, out);
}